// LAHGCN_28870770163985
// MI455X (gfx1250) — compile-verified
//
#include <hip/hip_runtime.h>

typedef __attribute__((ext_vector_type(2))) float v2f;
typedef __attribute__((ext_vector_type(8))) float v8f;

#define N_NODES 50000
#define N_EDGES 20000
#define NNZ     1600000
#define CONCAT  4
#define C_IN    256
#define C_HID   256
#define C_OUT   40
#define C_CAT   (CONCAT * C_HID)   // 1024
#define M_TILES (N_NODES / 16)     // 3125 (exact)

// ---- workspace layout (float offsets) ----
// dv_is [50000] | de_inv [20000] | H/S [50000*1024] | E region [20000*1024]
//   E region reused later: O [50000*40] at +0, E2 [20000*40] at +2,000,000
#define OFF_DVIS  ((size_t)0)
#define OFF_DEINV ((size_t)N_NODES)
#define OFF_H     ((size_t)(N_NODES + N_EDGES))                 // 70000 (16B aligned)
#define OFF_E     (OFF_H + (size_t)N_NODES * C_CAT)             // 51,270,000
#define OFF_O     OFF_E
#define OFF_E2    (OFF_E + (size_t)N_NODES * C_OUT)

// -------------------- utility --------------------
__global__ void k_zero(float* __restrict__ p, size_t n) {
  size_t i = (size_t)blockIdx.x * blockDim.x + threadIdx.x;
  if (i < n) p[i] = 0.0f;
}

// -------------------- degrees --------------------
__global__ void k_deg(const int* __restrict__ ni, const int* __restrict__ ei,
                      float* __restrict__ dv, float* __restrict__ de) {
  int i = blockIdx.x * blockDim.x + threadIdx.x;
  if (i < NNZ) {
    atomicAdd(&dv[ni[i]], 1.0f);
    atomicAdd(&de[ei[i]], 1.0f);
  }
}

__global__ void k_finalize_deg(float* __restrict__ dv, float* __restrict__ de) {
  int i = blockIdx.x * blockDim.x + threadIdx.x;
  if (i < N_NODES) {
    float d = dv[i];
    dv[i] = (d > 0.0f) ? rsqrtf(d) : 0.0f;
  }
  if (i < N_EDGES) {
    float d = de[i];
    de[i] = (d > 0.0f) ? (1.0f / d) : 0.0f;
  }
}

// -------------------- GEMM1: H[n, c*256+k] = x[c]@W1[c] + b1[c] --------------------
// One wave computes a 16(M) x 64(N) tile: 4 accumulators share one A fragment.
// f32 WMMA 16x16x4 fragment layout (ISA 7.12.2):
//   A: lane<16 -> M=lane, {K=k0,k0+1}; lane>=16 -> M=lane-16, {K=k0+2,k0+3}
//   B: mirrored (rows k0/k0+1 in lanes 0-15, rows k0+2/k0+3 in lanes 16-31)
//   C/D: vgpr r -> M = r + 8*(lane>=16), N = lane&15
__global__ __launch_bounds__(256)
void k_gemm1(const float* __restrict__ x, const float* __restrict__ W1,
             const float* __restrict__ b1, float* __restrict__ H) {
  int wave = blockIdx.x * (blockDim.x >> 5) + (threadIdx.x >> 5);
  if (wave >= CONCAT * M_TILES * 4) return;     // wave-uniform guard (EXEC stays all-1)
  int lane = threadIdx.x & 31;
  int half = lane >> 4;
  int l16  = lane & 15;

  int nq = wave & 3;                 // 64-column quad within branch
  int mt = (wave >> 2) % M_TILES;    // 16-row tile
  int c  = (wave >> 2) / M_TILES;    // branch

  const float* A = x + ((size_t)c * N_NODES + (size_t)mt * 16 + l16) * C_IN;
  const float* B = W1 + (size_t)c * C_IN * C_HID + (size_t)(nq * 64 + l16);

  v8f acc[4] = {};
#pragma unroll 4
  for (int k0 = 0; k0 < C_IN; k0 += 4) {
    int kb = k0 + half * 2;
    v2f a;
    a.x = A[kb];
    a.y = A[kb + 1];
    const float* Bk = B + (size_t)kb * C_HID;
#pragma unroll
    for (int j = 0; j < 4; ++j) {
      v2f b;
      b.x = Bk[j * 16];
      b.y = Bk[j * 16 + C_HID];
      acc[j] = __builtin_amdgcn_wmma_f32_16x16x4_f32(false, a, false, b,
                                                     (short)0, acc[j], false, false);
    }
  }

#pragma unroll
  for (int j = 0; j < 4; ++j) {
    int col = c * C_HID + nq * 64 + j * 16 + l16;
    float bias = b1[c * C_HID + nq * 64 + j * 16 + l16];
#pragma unroll
    for (int r = 0; r < 8; ++r) {
      size_t row = (size_t)mt * 16 + r + 8 * half;
      H[row * C_CAT + col] = acc[j][r] + bias;
    }
  }
}

// -------------------- smooth (1024 ch), hop 1: E[e] += H[v] * dv_is[v] --------------------
__global__ void k_scatter_ve(const int* __restrict__ ni, const int* __restrict__ ei,
                             const float* __restrict__ dvis,
                             const float* __restrict__ H, float* __restrict__ E) {
  unsigned idx = blockIdx.x * blockDim.x + threadIdx.x;   // NNZ * 256 float4-groups
  if (idx >= (unsigned)NNZ * (C_CAT / 4)) return;
  unsigned i  = idx >> 8;
  unsigned cg = idx & 255;
  int v = ni[i], e = ei[i];
  float s = dvis[v];
  float4 h = reinterpret_cast<const float4*>(H + (size_t)v * C_CAT)[cg];
  float* dst = E + (size_t)e * C_CAT + (size_t)cg * 4;
  atomicAdd(dst + 0, h.x * s);
  atomicAdd(dst + 1, h.y * s);
  atomicAdd(dst + 2, h.z * s);
  atomicAdd(dst + 3, h.w * s);
}

// hop 2: S[v] += E[e] * de_inv[e]
__global__ void k_gather_ev(const int* __restrict__ ni, const int* __restrict__ ei,
                            const float* __restrict__ deinv,
                            const float* __restrict__ E, float* __restrict__ S) {
  unsigned idx = blockIdx.x * blockDim.x + threadIdx.x;
  if (idx >= (unsigned)NNZ * (C_CAT / 4)) return;
  unsigned i  = idx >> 8;
  unsigned cg = idx & 255;
  int v = ni[i], e = ei[i];
  float s = deinv[e];
  float4 f = reinterpret_cast<const float4*>(E + (size_t)e * C_CAT)[cg];
  float* dst = S + (size_t)v * C_CAT + (size_t)cg * 4;
  atomicAdd(dst + 0, f.x * s);
  atomicAdd(dst + 1, f.y * s);
  atomicAdd(dst + 2, f.z * s);
  atomicAdd(dst + 3, f.w * s);
}

// S = relu(S * dv_is[v])
__global__ void k_relu_scale(float* __restrict__ S, const float* __restrict__ dvis) {
  size_t idx = (size_t)blockIdx.x * blockDim.x + threadIdx.x;
  if (idx >= (size_t)N_NODES * C_CAT) return;
  size_t v = idx >> 10;   // C_CAT == 1024
  float val = S[idx] * dvis[v];
  S[idx] = val > 0.0f ? val : 0.0f;
}

// -------------------- GEMM2: O = S[50000,1024] @ W2[1024,40] + b2 --------------------
__global__ __launch_bounds__(256)
void k_gemm2(const float* __restrict__ S, const float* __restrict__ W2,
             const float* __restrict__ b2, float* __restrict__ O) {
  int wave = blockIdx.x * (blockDim.x >> 5) + (threadIdx.x >> 5);
  if (wave >= M_TILES) return;                  // wave-uniform
  int lane = threadIdx.x & 31;
  int half = lane >> 4;
  int l16  = lane & 15;

  const float* A = S + ((size_t)wave * 16 + l16) * C_CAT;
  v8f acc[3] = {};
#pragma unroll 2
  for (int k0 = 0; k0 < C_CAT; k0 += 4) {
    int kb = k0 + half * 2;
    v2f a;
    a.x = A[kb];
    a.y = A[kb + 1];
#pragma unroll
    for (int jt = 0; jt < 3; ++jt) {
      int n = jt * 16 + l16;
      v2f b;
      b.x = (n < C_OUT) ? W2[(size_t)kb * C_OUT + n] : 0.0f;
      b.y = (n < C_OUT) ? W2[(size_t)(kb + 1) * C_OUT + n] : 0.0f;
      acc[jt] = __builtin_amdgcn_wmma_f32_16x16x4_f32(false, a, false, b,
                                                      (short)0, acc[jt], false, false);
    }
  }
  // stores only after all WMMAs; lane-divergent guard is safe here
#pragma unroll
  for (int jt = 0; jt < 3; ++jt) {
    int n = jt * 16 + l16;
    if (n < C_OUT) {
      float bias = b2[n];
#pragma unroll
      for (int r = 0; r < 8; ++r) {
        size_t row = (size_t)wave * 16 + r + 8 * half;
        O[row * C_OUT + n] = acc[jt][r] + bias;
      }
    }
  }
}

// -------------------- final smooth (40 ch) --------------------
__global__ void k_scatter40(const int* __restrict__ ni, const int* __restrict__ ei,
                            const float* __restrict__ dvis,
                            const float* __restrict__ O, float* __restrict__ E2) {
  unsigned idx = blockIdx.x * blockDim.x + threadIdx.x;
  if (idx >= (unsigned)NNZ * C_OUT) return;
  unsigned i = idx / C_OUT;
  unsigned ch = idx - i * C_OUT;
  int v = ni[i], e = ei[i];
  atomicAdd(&E2[(size_t)e * C_OUT + ch], O[(size_t)v * C_OUT + ch] * dvis[v]);
}

__global__ void k_gather40(const int* __restrict__ ni, const int* __restrict__ ei,
                           const float* __restrict__ deinv,
                           const float* __restrict__ E2, float* __restrict__ out) {
  unsigned idx = blockIdx.x * blockDim.x + threadIdx.x;
  if (idx >= (unsigned)NNZ * C_OUT) return;
  unsigned i = idx / C_OUT;
  unsigned ch = idx - i * C_OUT;
  int v = ni[i], e = ei[i];
  atomicAdd(&out[(size_t)v * C_OUT + ch], E2[(size_t)e * C_OUT + ch] * deinv[e]);
}

__global__ void k_scale40(float* __restrict__ out, const float* __restrict__ dvis) {
  unsigned idx = blockIdx.x * blockDim.x + threadIdx.x;
  if (idx >= (unsigned)N_NODES * C_OUT) return;
  out[idx] *= dvis[idx / C_OUT];
}

// -------------------- launcher --------------------
static inline unsigned nblk(size_t n, int t) { return (unsigned)((n + t - 1) / t); }

extern "C" void kernel_launch(void* const* d_in, const int* in_sizes, int n_in,
                              void* d_out, int out_size, void* d_ws, size_t ws_size,
                              hipStream_t stream) {
  const float* x  = (const float*)d_in[0];
  const float* W1 = (const float*)d_in[1];
  const float* b1 = (const float*)d_in[2];
  const float* W2 = (const float*)d_in[3];
  const float* b2 = (const float*)d_in[4];
  const int*   ni = (const int*)d_in[5];
  const int*   ei = (const int*)d_in[6];

  float* ws  = (float*)d_ws;
  float* dv  = ws + OFF_DVIS;   // becomes dv_is
  float* de  = ws + OFF_DEINV;  // becomes de_inv
  float* H   = ws + OFF_H;      // later reused as S
  float* E   = ws + OFF_E;
  float* O   = ws + OFF_O;      // aliases E (E dead by then)
  float* E2  = ws + OFF_E2;     // aliases E tail
  float* out = (float*)d_out;

  // degrees
  k_zero<<<nblk(N_NODES + N_EDGES, 256), 256, 0, stream>>>(dv, N_NODES + N_EDGES);
  k_deg<<<nblk(NNZ, 256), 256, 0, stream>>>(ni, ei, dv, de);
  k_finalize_deg<<<nblk(N_NODES, 256), 256, 0, stream>>>(dv, de);

  // branch GEMM into concatenated layout
  k_gemm1<<<nblk((size_t)CONCAT * M_TILES * 4 * 32, 256), 256, 0, stream>>>(x, W1, b1, H);

  // 1024-channel smooth (all 4 branches at once)
  k_zero<<<nblk((size_t)N_EDGES * C_CAT, 256), 256, 0, stream>>>(E, (size_t)N_EDGES * C_CAT);
  k_scatter_ve<<<nblk((size_t)NNZ * (C_CAT / 4), 256), 256, 0, stream>>>(ni, ei, dv, H, E);
  k_zero<<<nblk((size_t)N_NODES * C_CAT, 256), 256, 0, stream>>>(H, (size_t)N_NODES * C_CAT); // H -> S
  k_gather_ev<<<nblk((size_t)NNZ * (C_CAT / 4), 256), 256, 0, stream>>>(ni, ei, de, E, H);
  k_relu_scale<<<nblk((size_t)N_NODES * C_CAT, 256), 256, 0, stream>>>(H, dv);

  // output GEMM
  k_gemm2<<<nblk((size_t)M_TILES * 32, 256), 256, 0, stream>>>(H, W2, b2, O);

  // final 40-channel smooth
  k_zero<<<nblk((size_t)N_EDGES * C_OUT, 256), 256, 0, stream>>>(E2, (size_t)N_EDGES * C_OUT);
  k_zero<<<nblk((size_t)N_NODES * C_OUT, 256), 256, 0, stream>>>(out, (size_t)N_NODES * C_OUT);
  k_scatter40<<<nblk((size_t)NNZ * C_OUT, 256), 256, 0, stream>>>(ni, ei, dv, O, E2);
  k_gather40<<<nblk((size_t)NNZ * C_OUT, 256), 256, 0, stream>>>(ni, ei, de, E2, out);
  k_scale40<<<nblk((size_t)N_NODES * C_OUT, 256), 256, 0, stream>>>(out, dv);
}